// GINConvEncoder_36919538876779
// MI455X (gfx1250) — compile-verified
//
#include <hip/hip_runtime.h>

#define NNODES 50000
#define NEDGES 800000
#define DIM 128
#define NLAYERS 4
#define NGRAPHS 256

typedef float v2f __attribute__((ext_vector_type(2)));
typedef float v8f __attribute__((ext_vector_type(8)));

// Guaranteed native L2 fp32 atomic (non-returning, device scope).
// Tracked by STOREcnt; S_ENDPGM's implicit wait-idle covers completion.
__device__ __forceinline__ void atomAddF(float* p, float v) {
  asm volatile("global_atomic_add_f32 %0, %1, off scope:SCOPE_DEV"
               :: "v"(p), "v"(v)
               : "memory");
}

// ---------------------------------------------------------------------------
// init: h = x, agg = x, zero pooled-output + BN accumulators
// ---------------------------------------------------------------------------
__global__ __launch_bounds__(256) void init_k(const float* __restrict__ x,
                                              float* __restrict__ h,
                                              float* __restrict__ agg,
                                              float* __restrict__ zbuf,
                                              int nzero4) {
  int i = blockIdx.x * 256 + threadIdx.x;  // float4 index over NNODES*DIM/4
  float4 v = ((const float4*)x)[i];
  ((float4*)h)[i] = v;
  ((float4*)agg)[i] = v;
  if (i < nzero4) ((float4*)zbuf)[i] = make_float4(0.f, 0.f, 0.f, 0.f);
}

// ---------------------------------------------------------------------------
// edge scatter: agg[dst] += h[src]  (one wave per edge, float4 per lane)
// ---------------------------------------------------------------------------
__global__ __launch_bounds__(256) void edge_k(const float* __restrict__ h,
                                              float* __restrict__ agg,
                                              const long long* __restrict__ ei) {
  long long t = (long long)blockIdx.x * 256 + threadIdx.x;
  int e = (int)(t >> 5);
  int lane = (int)(t & 31);
  if (e >= NEDGES) return;
  int src = (int)ei[e];
  int dst = (int)ei[(size_t)NEDGES + e];
  float4 v = *(const float4*)(h + (size_t)src * DIM + lane * 4);
  float* p = agg + (size_t)dst * DIM + lane * 4;
  atomAddF(p + 0, v.x);
  atomAddF(p + 1, v.y);
  atomAddF(p + 2, v.z);
  atomAddF(p + 3, v.w);
}

// ---------------------------------------------------------------------------
// C(MxD) = relu(A(MxD) @ W(DxD) + bias)   via V_WMMA_F32_16X16X4_F32
// block = 256 thr (8 waves); block tile = 64 rows x 128 cols; wave = 16 cols.
// A tile staged in LDS, row stride 132 floats (stride%64==4 -> the wave32
// ds_load_b64 fragment read touches 64 distinct banks, conflict-free).
// ---------------------------------------------------------------------------
__global__ __launch_bounds__(256) void gemm_relu_k(const float* __restrict__ A,
                                                   const float* __restrict__ W,
                                                   const float* __restrict__ bias,
                                                   float* __restrict__ C, int M) {
  __shared__ float As[64][132];
  const int tid = threadIdx.x;
  const int wave = tid >> 5;
  const int lane = tid & 31;
  const int row0 = blockIdx.x * 64;

  // stage 64x128 A tile (zero-pad rows >= M)
#pragma unroll
  for (int i = 0; i < 8; ++i) {
    int idx = tid + i * 256;  // 2048 float4 in tile
    int r = idx >> 5;         // 32 float4 per row
    int c4 = idx & 31;
    float4 v = make_float4(0.f, 0.f, 0.f, 0.f);
    int row = row0 + r;
    if (row < M) v = *(const float4*)(A + (size_t)row * DIM + c4 * 4);
    float* p = &As[r][c4 * 4];
    p[0] = v.x; p[1] = v.y; p[2] = v.z; p[3] = v.w;
  }
  __syncthreads();

  const int nh = lane & 15;
  const int col = wave * 16 + nh;       // output column for this lane
  const int khalf = (lane >> 4) << 1;   // 0 (lanes 0-15) or 2 (lanes 16-31)
  const float bval = bias[col];

  v8f acc0 = {}, acc1 = {}, acc2 = {}, acc3 = {};

  for (int k = 0; k < DIM; k += 4) {
    // B fragment: per-lane holds W[k+khalf][col], W[k+khalf+1][col]
    v2f b;
    b.x = W[(size_t)(k + khalf) * DIM + col];
    b.y = W[(size_t)(k + khalf + 1) * DIM + col];
    // A fragments: per-lane row nh, K pair (khalf, khalf+1)
    v2f a0 = *(const v2f*)&As[nh][k + khalf];
    v2f a1 = *(const v2f*)&As[16 + nh][k + khalf];
    v2f a2 = *(const v2f*)&As[32 + nh][k + khalf];
    v2f a3 = *(const v2f*)&As[48 + nh][k + khalf];
    acc0 = __builtin_amdgcn_wmma_f32_16x16x4_f32(false, a0, false, b, (short)0, acc0, false, false);
    acc1 = __builtin_amdgcn_wmma_f32_16x16x4_f32(false, a1, false, b, (short)0, acc1, false, false);
    acc2 = __builtin_amdgcn_wmma_f32_16x16x4_f32(false, a2, false, b, (short)0, acc2, false, false);
    acc3 = __builtin_amdgcn_wmma_f32_16x16x4_f32(false, a3, false, b, (short)0, acc3, false, false);
  }

  // C/D layout: vgpr i -> M = i (lanes 0-15) or M = 8+i (lanes 16-31)
  const int rsel = (lane >> 4) << 3;
  if (row0 + 64 <= M) {
    // full tile: wave-uniform branch, unguarded stores
#pragma unroll
    for (int i = 0; i < 8; ++i) {
      C[(size_t)(row0 + rsel + i) * DIM + col]      = fmaxf(acc0[i] + bval, 0.f);
      C[(size_t)(row0 + 16 + rsel + i) * DIM + col] = fmaxf(acc1[i] + bval, 0.f);
      C[(size_t)(row0 + 32 + rsel + i) * DIM + col] = fmaxf(acc2[i] + bval, 0.f);
      C[(size_t)(row0 + 48 + rsel + i) * DIM + col] = fmaxf(acc3[i] + bval, 0.f);
    }
  } else {
#pragma unroll
    for (int i = 0; i < 8; ++i) {
      int r0 = row0 + rsel + i;
      if (r0 < M) C[(size_t)r0 * DIM + col] = fmaxf(acc0[i] + bval, 0.f);
      int r1 = row0 + 16 + rsel + i;
      if (r1 < M) C[(size_t)r1 * DIM + col] = fmaxf(acc1[i] + bval, 0.f);
      int r2 = row0 + 32 + rsel + i;
      if (r2 < M) C[(size_t)r2 * DIM + col] = fmaxf(acc2[i] + bval, 0.f);
      int r3 = row0 + 48 + rsel + i;
      if (r3 < M) C[(size_t)r3 * DIM + col] = fmaxf(acc3[i] + bval, 0.f);
    }
  }
}

// ---------------------------------------------------------------------------
// BN stats: per-column sum / sumsq partials (coalesced row reads)
// grid = 400 blocks x 128 thr; 125 rows per block (400*125 == 50000)
// ---------------------------------------------------------------------------
__global__ __launch_bounds__(128) void bn_stats_k(const float* __restrict__ h2,
                                                  float* __restrict__ gsum,
                                                  float* __restrict__ gsumsq) {
  int c = threadIdx.x;
  int r0 = blockIdx.x * 125;
  float s = 0.f, s2 = 0.f;
  for (int r = r0; r < r0 + 125; ++r) {
    float v = h2[(size_t)r * DIM + c];
    s += v;
    s2 = fmaf(v, v, s2);
  }
  atomAddF(&gsum[c], s);
  atomAddF(&gsumsq[c], s2);
}

__global__ void bn_final_k(float* gsum, float* gsumsq,
                           const float* __restrict__ gamma,
                           const float* __restrict__ beta,
                           float* scale, float* shift) {
  int c = threadIdx.x;
  const float invn = 1.f / (float)NNODES;
  float mu = gsum[c] * invn;
  float var = fmaxf(gsumsq[c] * invn - mu * mu, 0.f);
  float rs = rsqrtf(var + 1e-5f);
  float sc = rs * gamma[c];
  scale[c] = sc;
  shift[c] = fmaf(-mu, sc, beta[c]);
  gsum[c] = 0.f;   // ready for next layer
  gsumsq[c] = 0.f;
}

// h = agg = h2*scale + shift  (pre-seeds next layer's (1+eps)*h term)
__global__ __launch_bounds__(256) void bn_apply_k(float* __restrict__ h2agg,
                                                  float* __restrict__ h,
                                                  const float* __restrict__ scale,
                                                  const float* __restrict__ shift) {
  int i = blockIdx.x * 256 + threadIdx.x;  // float4 index
  int c4 = i & 31;
  float4 sc = ((const float4*)scale)[c4];
  float4 sh = ((const float4*)shift)[c4];
  float4 v = ((float4*)h2agg)[i];
  v.x = fmaf(v.x, sc.x, sh.x);
  v.y = fmaf(v.y, sc.y, sh.y);
  v.z = fmaf(v.z, sc.z, sh.z);
  v.w = fmaf(v.w, sc.w, sh.w);
  ((float4*)h2agg)[i] = v;
  ((float4*)h)[i] = v;
}

// ---------------------------------------------------------------------------
// global_add_pool: batch_idx is sorted -> run-length accumulate, 1 atomic per
// segment boundary per column. block = 128 thr (one column each), 16 nodes.
// ---------------------------------------------------------------------------
__global__ __launch_bounds__(128) void pool_k(const float* __restrict__ h,
                                              const long long* __restrict__ batch,
                                              float* __restrict__ g) {
  int c = threadIdx.x;
  int base = blockIdx.x * 16;
  float acc = 0.f;
  int cur = -1;
  for (int j = 0; j < 16; ++j) {
    int node = base + j;
    if (node >= NNODES) break;
    int bi = (int)batch[node];
    if (bi != cur) {
      if (cur >= 0) atomAddF(&g[(size_t)cur * DIM + c], acc);
      cur = bi;
      acc = 0.f;
    }
    acc += h[(size_t)node * DIM + c];
  }
  if (cur >= 0) atomAddF(&g[(size_t)cur * DIM + c], acc);
}

// ---------------------------------------------------------------------------
extern "C" void kernel_launch(void* const* d_in, const int* in_sizes, int n_in,
                              void* d_out, int out_size, void* d_ws, size_t ws_size,
                              hipStream_t stream) {
  const float* x        = (const float*)d_in[0];
  const long long* ei   = (const long long*)d_in[1];
  const long long* bidx = (const long long*)d_in[2];
  const float* W1    = (const float*)d_in[3];
  const float* b1    = (const float*)d_in[4];
  const float* W2    = (const float*)d_in[5];
  const float* b2    = (const float*)d_in[6];
  const float* gamma = (const float*)d_in[7];
  const float* beta  = (const float*)d_in[8];
  const float* Wp    = (const float*)d_in[9];
  const float* bp    = (const float*)d_in[10];
  float* out = (float*)d_out;

  float* ws = (float*)d_ws;
  const size_t NND = (size_t)NNODES * DIM;  // 6.4M floats
  float* h      = ws;
  float* agg    = ws + NND;
  float* tmp    = ws + 2 * NND;
  float* g      = ws + 3 * NND;                 // 256*128
  float* gsum   = g + (size_t)NGRAPHS * DIM;    // 128
  float* gsumsq = gsum + DIM;                   // 128
  float* scale  = gsumsq + DIM;                 // 128
  float* shift  = scale + DIM;                  // 128

  const int nzero4 = (NGRAPHS * DIM + 4 * DIM) / 4;  // g + 4 small vectors
  const int elem4_grid = (int)(NND / 4 / 256);       // 6250
  init_k<<<elem4_grid, 256, 0, stream>>>(x, h, agg, g, nzero4);

  const int gemm_grid = (NNODES + 63) / 64;          // 782
  const int edge_grid = (NEDGES * 32) / 256;         // 100000

  for (int l = 0; l < NLAYERS; ++l) {
    edge_k<<<edge_grid, 256, 0, stream>>>(h, agg, ei);
    gemm_relu_k<<<gemm_grid, 256, 0, stream>>>(agg, W1 + (size_t)l * DIM * DIM,
                                               b1 + (size_t)l * DIM, tmp, NNODES);
    gemm_relu_k<<<gemm_grid, 256, 0, stream>>>(tmp, W2 + (size_t)l * DIM * DIM,
                                               b2 + (size_t)l * DIM, agg, NNODES);
    bn_stats_k<<<400, 128, 0, stream>>>(agg, gsum, gsumsq);
    bn_final_k<<<1, 128, 0, stream>>>(gsum, gsumsq, gamma + (size_t)l * DIM,
                                      beta + (size_t)l * DIM, scale, shift);
    bn_apply_k<<<elem4_grid, 256, 0, stream>>>(agg, h, scale, shift);
  }

  pool_k<<<(NNODES + 15) / 16, 128, 0, stream>>>(h, bidx, g);
  gemm_relu_k<<<(NGRAPHS + 63) / 64, 256, 0, stream>>>(g, Wp, bp, out, NGRAPHS);
}